// Model_73065983640004
// MI455X (gfx1250) — compile-verified
//
#include <hip/hip_runtime.h>

#define D 64

typedef __attribute__((ext_vector_type(2))) float v2f;
typedef __attribute__((ext_vector_type(8))) float v8f;

// ---------------------------------------------------------------------------
// One wave per edge. Lanes 0..31 each carry 2 of the 64 feature dims (float2).
//   iB[ei] += w * uA[eu]   and   uB[eu] += w * iA[ei]
// Edge scalars are wave-uniform -> scalar loads. Row loads are 256B coalesced.
// Atomic destinations (77MB total) live in the 192MB L2, so the 128 f32
// atomics/edge resolve at L2 bank rate.
// ---------------------------------------------------------------------------
__global__ void __launch_bounds__(256) scatter_edges(
    const float* __restrict__ uA, const float* __restrict__ iA,
    float* __restrict__ uB, float* __restrict__ iB,
    const int* __restrict__ edge_u, const int* __restrict__ edge_i,
    const float* __restrict__ norm, int n_edges)
{
    int wave = (int)(blockIdx.x * (blockDim.x >> 5) + (threadIdx.x >> 5));
    int lane = threadIdx.x & 31;
    if (wave >= n_edges) return;

    int   u  = edge_u[wave];
    int   it = edge_i[wave];
    float w  = norm[wave];

    const v2f* us = (const v2f*)(uA + (size_t)u  * D);
    const v2f* is = (const v2f*)(iA + (size_t)it * D);
    v2f uv = us[lane];
    v2f iv = is[lane];

    float* id = iB + (size_t)it * D + lane * 2;
    float* ud = uB + (size_t)u  * D + lane * 2;
    atomicAdd(&id[0], uv.x * w);
    atomicAdd(&id[1], uv.y * w);
    atomicAdd(&ud[0], iv.x * w);
    atomicAdd(&ud[1], iv.y * w);
}

// ---------------------------------------------------------------------------
// Gather rows at the 3*batch requested indices into d_out (set or +=).
// out rows: [0,B) = user table[users], [B,2B) = item[pos], [2B,3B) = item[neg]
// ---------------------------------------------------------------------------
__global__ void __launch_bounds__(256) gather_rows(
    const float* __restrict__ utab, const float* __restrict__ itab,
    const int* __restrict__ users, const int* __restrict__ pos,
    const int* __restrict__ neg,
    float* __restrict__ out, int batch, int accumulate)
{
    int gid = blockIdx.x * blockDim.x + threadIdx.x;
    int total = 3 * batch * D;
    if (gid >= total) return;

    int d   = gid & (D - 1);
    int r   = (gid / D) % batch;
    int seg = gid / (batch * D);

    const float* src;
    if (seg == 0)      src = utab + (size_t)users[r] * D;
    else if (seg == 1) src = itab + (size_t)pos[r]   * D;
    else               src = itab + (size_t)neg[r]   * D;

    float v = src[d];
    if (accumulate) out[gid] += v;
    else            out[gid]  = v;
}

// ---------------------------------------------------------------------------
// Final scale by 1/(NUM_LAYERS+1) = 0.25 via V_WMMA_F32_16X16X4_F32:
//   D(16x16) = sum_{c=0..3} A_c(16x4) x B_c(4x16),  A = data tile, B = 0.25*I
// A-operand 16x4 f32 layout (documented): VGPR0: lanes0-15 K=0, lanes16-31 K=2;
// VGPR1: K=1 / K=3, M = lane&15. C/D: VGPR j holds M=j (lanes0-15) / M=j+8,
// N = lane&15. Power-of-two scale with identity B => bit-exact result.
// One wave per 16x16 tile; all 32 lanes active (EXEC all-ones) around WMMA.
// ---------------------------------------------------------------------------
__global__ void __launch_bounds__(256) wmma_scale_inplace(
    float* __restrict__ out, int ntiles)
{
    int wid = (int)(blockIdx.x * (blockDim.x >> 5) + (threadIdx.x >> 5));
    if (wid >= ntiles) return;            // wave-uniform exit only

    int lane = threadIdx.x & 31;
    int h = lane >> 4;                    // half: 0 -> lanes 0-15, 1 -> 16-31
    int l = lane & 15;                    // M (for A), N (for B/D)
    int rg = wid >> 2;                    // 16-row group
    int ct = wid & 3;                     // 16-col tile of the 64 dims

    float* base = out + (size_t)rg * 16 * D + ct * 16;

    v8f acc = {};
#pragma unroll
    for (int c = 0; c < 4; ++c) {
        int k0 = 2 * h;                   // K held in VGPR0 for this lane
        int k1 = 2 * h + 1;               // K held in VGPR1
        v2f a, b;
        a.x = base[(size_t)l * D + 4 * c + k0];     // A[m=l][k]
        a.y = base[(size_t)l * D + 4 * c + k1];
        b.x = ((4 * c + k0) == l) ? 0.25f : 0.0f;   // B[k][n=l] = 0.25*I
        b.y = ((4 * c + k1) == l) ? 0.25f : 0.0f;
        acc = __builtin_amdgcn_wmma_f32_16x16x4_f32(
            /*neg_a=*/false, a, /*neg_b=*/false, b,
            /*c_mod=*/(short)0, acc, /*reuse_a=*/false, /*reuse_b=*/false);
    }
#pragma unroll
    for (int j = 0; j < 8; ++j)
        base[(size_t)(h * 8 + j) * D + l] = acc[j];
}

// ---------------------------------------------------------------------------
// Orchestration: ping-pong u/i tables in d_ws; per-layer zero + scatter +
// gather-add into d_out; final WMMA scale. All ops are graph-capturable.
// ---------------------------------------------------------------------------
extern "C" void kernel_launch(void* const* d_in, const int* in_sizes, int n_in,
                              void* d_out, int out_size, void* d_ws, size_t ws_size,
                              hipStream_t stream)
{
    const float* user_feat = (const float*)d_in[0];
    const float* item_feat = (const float*)d_in[1];
    const int*   edge_u    = (const int*)d_in[2];
    const int*   edge_i    = (const int*)d_in[3];
    const float* norm      = (const float*)d_in[4];
    const int*   users     = (const int*)d_in[5];
    const int*   pos       = (const int*)d_in[6];
    const int*   neg       = (const int*)d_in[7];
    float* out = (float*)d_out;

    const int n_users = in_sizes[0] / D;
    const int n_items = in_sizes[1] / D;
    const int n_edges = in_sizes[2];
    const int batch   = in_sizes[5];

    size_t uElems = (size_t)n_users * D;   // 12.8M floats = 51.2 MB
    size_t iElems = (size_t)n_items * D;   //  6.4M floats = 25.6 MB

    float* u0 = (float*)d_ws;
    float* u1 = u0 + uElems;
    float* i0 = u1 + uElems;
    float* i1 = i0 + iElems;               // total ws use: 153.6 MB

    float* uCur = u0; float* uNxt = u1;
    float* iCur = i0; float* iNxt = i1;

    // h0 = input features
    hipMemcpyAsync(uCur, user_feat, uElems * sizeof(float),
                   hipMemcpyDeviceToDevice, stream);
    hipMemcpyAsync(iCur, item_feat, iElems * sizeof(float),
                   hipMemcpyDeviceToDevice, stream);

    // out = layer-0 contribution (raw features at requested rows)
    int gthreads = 3 * batch * D;
    int gblocks  = (gthreads + 255) / 256;
    gather_rows<<<gblocks, 256, 0, stream>>>(user_feat, item_feat,
                                             users, pos, neg, out, batch, 0);

    int sblocks = (n_edges + 7) / 8;       // 8 waves (edges) per 256-thr block
    for (int L = 0; L < 3; ++L) {
        hipMemsetAsync(uNxt, 0, uElems * sizeof(float), stream);
        hipMemsetAsync(iNxt, 0, iElems * sizeof(float), stream);
        scatter_edges<<<sblocks, 256, 0, stream>>>(uCur, iCur, uNxt, iNxt,
                                                   edge_u, edge_i, norm, n_edges);
        gather_rows<<<gblocks, 256, 0, stream>>>(uNxt, iNxt,
                                                 users, pos, neg, out, batch, 1);
        float* t;
        t = uCur; uCur = uNxt; uNxt = t;
        t = iCur; iCur = iNxt; iNxt = t;
    }

    // out *= 0.25 via v_wmma_f32_16x16x4_f32 (one wave per 16x16 tile)
    int nrows  = 3 * batch;                // 24576, divisible by 16
    int ntiles = (nrows / 16) * (D / 16);  // 6144 tiles
    int wblocks = (ntiles + 7) / 8;
    wmma_scale_inplace<<<wblocks, 256, 0, stream>>>(out, ntiles);
}